// ScaledDotProductAttention_61873298866888
// MI455X (gfx1250) — compile-verified
//
#include <hip/hip_runtime.h>
#include <hip/hip_bf16.h>

// Scaled dot-product attention forward for MI455X (gfx1250, wave32).
// Inputs (setup_inputs order): value(B,S,D) f32, key(B,S,D) f32,
// query(B,Q,D) f32, mask(B,Q,S) bool(uint8).  Output: out(B,Q,D) f32
// followed by attention(B,Q,S) f32, concatenated flat in d_out.

typedef __attribute__((ext_vector_type(2))) float v2f;
typedef __attribute__((ext_vector_type(8))) float v8f;

#define B_DIM 16
#define S_DIM 2048
#define Q_DIM 2048
#define D_DIM 64
#define SCALE 0.125f          // 64^-0.5
#define ROWS  16              // query rows per workgroup
#define SPAD  2052            // padded LDS row stride (floats): 16B-aligned, bank-skewed

__global__ __launch_bounds__(256)
void attn_fwd_kernel(const float* __restrict__ value,
                     const float* __restrict__ key,
                     const float* __restrict__ query,
                     const unsigned char* __restrict__ mask,
                     float* __restrict__ outp,
                     float* __restrict__ attn) {
    // ~140 KB LDS (gfx1250: up to 320 KB per workgroup)
    __shared__ float sS[ROWS][SPAD];      // score stripe 16 x 2048 (padded)
    __shared__ float sRed[ROWS][16];      // per-row partial reductions
    __shared__ float sRowMax[ROWS];
    __shared__ float sRowInv[ROWS];
    __shared__ float sOut[8][ROWS][16];   // per-wave P*V partials

    const int tid  = threadIdx.x;
    const int wave = tid >> 5;            // 8 waves (wave32)
    const int lane = tid & 31;
    const int b    = blockIdx.x >> 7;     // 128 q-tiles per batch
    const int q0   = (blockIdx.x & 127) << 4;

    // 32-bit WMMA 16x4 fragment layout: lanes 0-15 hold K=0,1; lanes 16-31 K=2,3
    const int rc   = lane & 15;           // row (A) / column (B, C) index
    const int koff = (lane >> 4) << 1;    // 0 or 2

    // ---------------- Phase 1: scores = Q K^T  (16 x 2048 into LDS) ----------
    v2f aq[16];
    {
        const float* qrow = query + ((size_t)b * Q_DIM + (q0 + rc)) * D_DIM;
#pragma unroll
        for (int kk = 0; kk < 16; ++kk)
            aq[kk] = *(const v2f*)(qrow + (kk << 2) + koff);
    }
    for (int st = wave; st < S_DIM / 16; st += 8) {
        const int s0 = st << 4;
        const float* krow = key + ((size_t)b * S_DIM + (s0 + rc)) * D_DIM;
        v8f c = {0.f, 0.f, 0.f, 0.f, 0.f, 0.f, 0.f, 0.f};
#pragma unroll
        for (int kk = 0; kk < 16; ++kk) {
            v2f bk = *(const v2f*)(krow + (kk << 2) + koff);
            c = __builtin_amdgcn_wmma_f32_16x16x4_f32(false, aq[kk], false, bk,
                                                      (short)0, c, false, false);
        }
        // C layout: VGPR j holds M=j (lanes 0-15) / M=j+8 (lanes 16-31), N=lane%16
        const int mbase = (lane >> 4) << 3;
#pragma unroll
        for (int j = 0; j < 8; ++j)
            sS[mbase + j][s0 + rc] = c[j] * SCALE;
    }
    __syncthreads();

    // ---------------- Phase 1.5: mask (True -> -inf), coalesced -------------
    {
        const unsigned char* mrow = mask + ((size_t)b * Q_DIM + q0) * S_DIM;
        const float NEG_INF = -__builtin_inff();
#pragma unroll
        for (int i = 0; i < (ROWS * S_DIM / 4) / 256; ++i) {  // 32 iters
            int idx4 = (i << 8) + tid;        // 4-byte group index
            int r = idx4 >> 9;                // 512 groups per row
            int s = (idx4 & 511) << 2;
            uchar4 mv = *(const uchar4*)(mrow + (size_t)r * S_DIM + s);
            if (mv.x) sS[r][s + 0] = NEG_INF;
            if (mv.y) sS[r][s + 1] = NEG_INF;
            if (mv.z) sS[r][s + 2] = NEG_INF;
            if (mv.w) sS[r][s + 3] = NEG_INF;
        }
    }
    __syncthreads();

    // ---------------- Phase 2: softmax (16 threads per row) ------------------
    const int r2 = tid >> 4;
    const int c2 = tid & 15;
    {
        float mx = -__builtin_inff();
        for (int i = 0; i < S_DIM / 16; ++i)
            mx = fmaxf(mx, sS[r2][c2 + (i << 4)]);
        sRed[r2][c2] = mx;
    }
    __syncthreads();
    if (tid < ROWS) {
        float mx = sRed[tid][0];
#pragma unroll
        for (int k = 1; k < 16; ++k) mx = fmaxf(mx, sRed[tid][k]);
        sRowMax[tid] = mx;
    }
    __syncthreads();
    {
        const float rm = sRowMax[r2];
        float sum = 0.f;
        for (int i = 0; i < S_DIM / 16; ++i) {
            int cc = c2 + (i << 4);
            float e = __expf(sS[r2][cc] - rm);
            sS[r2][cc] = e;                 // keep unnormalized P for P*V
            sum += e;
        }
        sRed[r2][c2] = sum;
    }
    __syncthreads();
    if (tid < ROWS) {
        float s = 0.f;
#pragma unroll
        for (int k = 0; k < 16; ++k) s += sRed[tid][k];
        sRowInv[tid] = 1.f / s;
    }
    __syncthreads();

    // ---------------- Phase 2c: write normalized attention (float4) ----------
    {
        float* arow = attn + ((size_t)b * Q_DIM + q0) * S_DIM;
#pragma unroll 4
        for (int i = 0; i < (ROWS * S_DIM / 4) / 256; ++i) {  // 32 iters
            int idx4 = (i << 8) + tid;
            int r  = idx4 >> 9;
            int cc = (idx4 & 511) << 2;
            float inv = sRowInv[r];
            float4 p = *(const float4*)&sS[r][cc];
            p.x *= inv; p.y *= inv; p.z *= inv; p.w *= inv;
            *(float4*)(arow + (size_t)r * S_DIM + cc) = p;
        }
    }

    // ---------------- Phase 3: out = P * V (then scale by 1/sum) -------------
    {
        const int nt = wave & 3;              // D-tile (16 cols)
        const int sh = wave >> 2;             // S-half (1024)
        const int dc = nt << 4;
        const float* vcol = value + (size_t)b * S_DIM * D_DIM + dc + rc;
        v8f acc = {0.f, 0.f, 0.f, 0.f, 0.f, 0.f, 0.f, 0.f};
        const int sbeg = sh << 10;
        for (int sb = sbeg; sb < sbeg + 1024; sb += 4) {
            v2f a = *(const v2f*)&sS[rc][sb + koff];          // P 16x4 chunk
            v2f bv;
            bv.x = vcol[(size_t)(sb + koff)     * D_DIM];     // V 4x16 chunk
            bv.y = vcol[(size_t)(sb + koff + 1) * D_DIM];
            acc = __builtin_amdgcn_wmma_f32_16x16x4_f32(false, a, false, bv,
                                                        (short)0, acc, false, false);
        }
        const int mbase = (lane >> 4) << 3;
#pragma unroll
        for (int j = 0; j < 8; ++j)
            sOut[wave][mbase + j][rc] = acc[j];
    }
    __syncthreads();
    {
        float* orow = outp + ((size_t)b * Q_DIM + q0) * D_DIM;
        for (int idx = tid; idx < ROWS * D_DIM; idx += 256) {
            int m   = idx >> 6;
            int dcl = idx & 63;
            int nt2 = dcl >> 4;
            int nl  = dcl & 15;
            orow[(size_t)m * D_DIM + dcl] =
                (sOut[nt2][m][nl] + sOut[nt2 + 4][m][nl]) * sRowInv[m];
        }
    }
}

extern "C" void kernel_launch(void* const* d_in, const int* in_sizes, int n_in,
                              void* d_out, int out_size, void* d_ws, size_t ws_size,
                              hipStream_t stream) {
    const float*         value = (const float*)d_in[0];
    const float*         key   = (const float*)d_in[1];
    const float*         query = (const float*)d_in[2];
    const unsigned char* mask  = (const unsigned char*)d_in[3];  // jnp.bool_ = 1 byte

    float* outp = (float*)d_out;
    float* attn = outp + (size_t)B_DIM * Q_DIM * D_DIM;

    dim3 grid(B_DIM * (Q_DIM / 16));   // 2048 blocks, one 16-row q-tile each
    attn_fwd_kernel<<<grid, 256, 0, stream>>>(value, key, query, mask, outp, attn);
}